// GNNLayer_79053168050534
// MI455X (gfx1250) — compile-verified
//
#include <hip/hip_runtime.h>
#include <math.h>

typedef __attribute__((ext_vector_type(2))) float v2f;
typedef __attribute__((ext_vector_type(8))) float v8f;

#define IN_DIM    128
#define OUT_DIM   128
#define ATTN_DIM  64
#define N_REL_TOT 401     // 2*N_REL+1
#define N_NODE    50000
#define N_EDGE    500000
#define N_QUERY   64
#define N_DT      731     // delta_tau in [-365, 365]

// ---------------- Precompute kernels (tiny, VALU) ----------------

// qpre[q][a] = rela_embed[q_rel[q]] @ Wqr_W + Wqr_b   (64 x 64)
__global__ void qpre_kernel(const int* __restrict__ q_rel,
                            const float* __restrict__ rela,
                            const float* __restrict__ WqrW,
                            const float* __restrict__ Wqrb,
                            float* __restrict__ qpre) {
    int q = blockIdx.x, a = threadIdx.x;
    const float* h = rela + q_rel[q] * IN_DIM;
    float s = Wqrb[a];
    for (int k = 0; k < IN_DIM; ++k) s += h[k] * WqrW[k * ATTN_DIM + a];
    qpre[q * ATTN_DIM + a] = s;
}

// rpre[r][a] = rela_embed[r] @ Wr   (401 x 64)
__global__ void rpre_kernel(const float* __restrict__ rela,
                            const float* __restrict__ Wr,
                            float* __restrict__ rpre) {
    int r = blockIdx.x, a = threadIdx.x;
    const float* h = rela + r * IN_DIM;
    float s = 0.f;
    for (int k = 0; k < IN_DIM; ++k) s += h[k] * Wr[k * ATTN_DIM + a];
    rpre[r * ATTN_DIM + a] = s;
}

// htau_tab[t][c] = wt1[c]*dt + bt1[c] + sin(wt2[c]*dt + bt2[c]),  dt = t - 365
__global__ void htau_kernel(const float* __restrict__ wt1, const float* __restrict__ bt1,
                            const float* __restrict__ wt2, const float* __restrict__ bt2,
                            float* __restrict__ htau_tab) {
    int t = blockIdx.x, c = threadIdx.x;
    float dt = (float)t - 365.0f;
    htau_tab[t * IN_DIM + c] = wt1[c] * dt + bt1[c] + sinf(wt2[c] * dt + bt2[c]);
}

// htpre[t][a] = htau_tab[t] @ Wtau   (731 x 64)
__global__ void htpre_kernel(const float* __restrict__ htau_tab,
                             const float* __restrict__ Wtau,
                             float* __restrict__ htpre) {
    int t = blockIdx.x, a = threadIdx.x;
    const float* h = htau_tab + t * IN_DIM;
    float s = 0.f;
    for (int k = 0; k < IN_DIM; ++k) s += h[k] * Wtau[k * ATTN_DIM + a];
    htpre[t * ATTN_DIM + a] = s;
}

// ---------------- Edge kernel: WMMA hs@Ws + attention + scatter ----------------
// One wave (32 lanes) processes a tile of 16 edges. N_EDGE = 16 * 31250 exactly.
// f32 WMMA 16x16x4 layouts (wave32):
//   A : lane l -> M = l%16, VGPR v -> K = 4k + 2*(l/16) + v
//   B : lane l -> N = l%16, VGPR v -> K = 4k + 2*(l/16) + v
//   C : lane l, VGPR j -> M = j + 8*(l/16), N = l%16
__global__ __launch_bounds__(256)
void edge_kernel(const float* __restrict__ hidden,
                 const int*   __restrict__ edges,
                 const int*   __restrict__ q_tau,
                 const float* __restrict__ rela,
                 const float* __restrict__ Ws,
                 const float* __restrict__ waW,
                 const float* __restrict__ wab_p,
                 const float* __restrict__ qpre,
                 const float* __restrict__ rpre,
                 const float* __restrict__ htau_tab,
                 const float* __restrict__ htpre,
                 float*       __restrict__ msg) {
    __shared__ float sWs[32 * 4 * 32 * 2];   // 8192 f32 = 32KB, B-fragment order
    __shared__ float swa[ATTN_DIM];

    int tid = threadIdx.x;
    // Pack Ws[128][64] into WMMA B-fragment layout: ((k*4+n)*32 + lane)*2 + v
    for (int i = tid; i < 8192; i += 256) {
        int v = i & 1, lane = (i >> 1) & 31, n = (i >> 6) & 3, k = i >> 8;
        int K = 4 * k + 2 * (lane >> 4) + v;
        int N = 16 * n + (lane & 15);
        sWs[i] = Ws[K * ATTN_DIM + N];
    }
    if (tid < ATTN_DIM) swa[tid] = waW[tid];
    __syncthreads();

    int wave = tid >> 5, lane = tid & 31;
    int tile = blockIdx.x * 8 + wave;
    if (tile >= N_EDGE / 16) return;          // wave-uniform exit, EXEC stays full

    int row = lane & 15, lh = lane >> 4;

    // Every lane loads metadata for edge `row` (lanes r and r+16 both hold row r).
    const int* ep = edges + (tile * 16 + row) * 7;
    int r_idx = ep[0], rel = ep[2], tau = ep[4], sub = ep[5], obj = ep[6];
    int tq = q_tau[r_idx];
    if (tau < 0) tau = tq;
    int dtidx = tau - tq + 365;               // [0, 730]
    int pk1 = rel | (r_idx << 10) | (dtidx << 16);
    int pk2 = sub | (obj << 16);              // both < 65536

    // ---- hs @ Ws via 128 f32 WMMAs per tile ----
    const float* hs = hidden + (long)sub * IN_DIM;
    v8f acc0 = {}, acc1 = {}, acc2 = {}, acc3 = {};
    #pragma unroll
    for (int k = 0; k < 32; ++k) {
        v2f a = *(const v2f*)(hs + 4 * k + 2 * lh);
        const v2f* bp = (const v2f*)sWs + (k * 4) * 32 + lane;
        acc0 = __builtin_amdgcn_wmma_f32_16x16x4_f32(false, a, false, bp[0],   (short)0, acc0, false, false);
        acc1 = __builtin_amdgcn_wmma_f32_16x16x4_f32(false, a, false, bp[32],  (short)0, acc1, false, false);
        acc2 = __builtin_amdgcn_wmma_f32_16x16x4_f32(false, a, false, bp[64],  (short)0, acc2, false, false);
        acc3 = __builtin_amdgcn_wmma_f32_16x16x4_f32(false, a, false, bp[96],  (short)0, acc3, false, false);
    }

    // ---- bias tables, relu, alpha = sigmoid(relu(pre) . w_alpha + b) ----
    float wab = wab_p[0];
    float wa0 = swa[row], wa1 = swa[16 + row], wa2 = swa[32 + row], wa3 = swa[48 + row];
    float alpha[8];
    int subj[8], objj[8], relj[8], dtij[8];
    #pragma unroll
    for (int j = 0; j < 8; ++j) {
        int src = j + 8 * lh;                 // C row this vgpr-slot holds
        int m1 = __shfl(pk1, src);
        int m2 = __shfl(pk2, src);
        int rl  = m1 & 1023;
        int rq  = (m1 >> 10) & 63;
        int dti = (m1 >> 16) & 0xffff;
        relj[j] = rl; dtij[j] = dti;
        subj[j] = m2 & 0xffff; objj[j] = (m2 >> 16) & 0xffff;

        const float* rp = rpre  + rl  * ATTN_DIM;
        const float* qp = qpre  + rq  * ATTN_DIM;
        const float* tp = htpre + dti * ATTN_DIM;
        float p, part = 0.f;
        p = acc0[j] + rp[row]      + qp[row]      + tp[row];      p = fmaxf(p, 0.f); part += p * wa0;
        p = acc1[j] + rp[16 + row] + qp[16 + row] + tp[16 + row]; p = fmaxf(p, 0.f); part += p * wa1;
        p = acc2[j] + rp[32 + row] + qp[32 + row] + tp[32 + row]; p = fmaxf(p, 0.f); part += p * wa2;
        p = acc3[j] + rp[48 + row] + qp[48 + row] + tp[48 + row]; p = fmaxf(p, 0.f); part += p * wa3;
        // reduce the 64-wide dot across the 16 lanes of this half
        part += __shfl_xor(part, 1, 16);
        part += __shfl_xor(part, 2, 16);
        part += __shfl_xor(part, 4, 16);
        part += __shfl_xor(part, 8, 16);
        alpha[j] = 1.0f / (1.0f + expf(-(part + wab)));
    }

    // ---- scatter: msg_agg[obj] += alpha * (hs + hr + htau) ----
    // Half `lh` covers rows j+8*lh; its 16 lanes tile the 128 columns (coalesced).
    #pragma unroll
    for (int j = 0; j < 8; ++j) {
        const float* hsr = hidden   + (long)subj[j] * IN_DIM;
        const float* hrr = rela     + relj[j] * IN_DIM;
        const float* htr = htau_tab + dtij[j] * IN_DIM;
        float* dst = msg + (long)objj[j] * IN_DIM;
        float al = alpha[j];
        #pragma unroll
        for (int c = 0; c < 8; ++c) {
            int col = row + 16 * c;
            atomicAdd(dst + col, al * (hsr[col] + hrr[col] + htr[col]));
        }
    }
}

// ---------------- Final GEMM: out = msg_agg @ Wh  (50000x128 @ 128x128) ----------------
__global__ __launch_bounds__(256)
void out_gemm(const float* __restrict__ msg,
              const float* __restrict__ Wh,
              float* __restrict__ out) {
    __shared__ float sW[32 * 8 * 32 * 2];    // 16384 f32 = 64KB, B-fragment order
    int tid = threadIdx.x;
    for (int i = tid; i < 16384; i += 256) {
        int v = i & 1, lane = (i >> 1) & 31, n = (i >> 6) & 7, k = i >> 9;
        int K = 4 * k + 2 * (lane >> 4) + v;
        int N = 16 * n + (lane & 15);
        sW[i] = Wh[K * OUT_DIM + N];
    }
    __syncthreads();

    int wave = tid >> 5, lane = tid & 31;
    int strip = blockIdx.x * 8 + wave;       // 16-row strip; 50000 = 16*3125 exactly
    if (strip >= N_NODE / 16) return;
    int row = lane & 15, lh = lane >> 4;

    const float* ap = msg + (long)(strip * 16 + row) * IN_DIM;
    v8f acc[8] = {};
    #pragma unroll
    for (int k = 0; k < 32; ++k) {
        v2f a = *(const v2f*)(ap + 4 * k + 2 * lh);
        const v2f* bp = (const v2f*)sW + (k * 8) * 32 + lane;
        #pragma unroll
        for (int n = 0; n < 8; ++n)
            acc[n] = __builtin_amdgcn_wmma_f32_16x16x4_f32(false, a, false, bp[n * 32],
                                                           (short)0, acc[n], false, false);
    }
    #pragma unroll
    for (int n = 0; n < 8; ++n)
        #pragma unroll
        for (int j = 0; j < 8; ++j)
            out[(long)(strip * 16 + j + 8 * lh) * OUT_DIM + 16 * n + row] = acc[n][j];
}

// ---------------- Host launcher ----------------
extern "C" void kernel_launch(void* const* d_in, const int* in_sizes, int n_in,
                              void* d_out, int out_size, void* d_ws, size_t ws_size,
                              hipStream_t stream) {
    const int*   q_rel  = (const int*)d_in[1];
    const int*   q_tau  = (const int*)d_in[2];
    const float* hidden = (const float*)d_in[3];
    const int*   edges  = (const int*)d_in[4];
    const float* rela   = (const float*)d_in[7];
    const float* Ws     = (const float*)d_in[8];
    const float* Wr     = (const float*)d_in[9];
    const float* WqrW   = (const float*)d_in[10];
    const float* Wqrb   = (const float*)d_in[11];
    const float* Wtau   = (const float*)d_in[12];
    const float* waW    = (const float*)d_in[13];
    const float* wab    = (const float*)d_in[14];
    const float* Wh     = (const float*)d_in[15];
    const float* wt1    = (const float*)d_in[16];
    const float* bt1    = (const float*)d_in[17];
    const float* wt2    = (const float*)d_in[18];
    const float* bt2    = (const float*)d_in[19];

    float* ws    = (float*)d_ws;
    float* msg   = ws;                                        // 50000*128
    float* qpre  = msg   + (size_t)N_NODE * IN_DIM;           // 64*64
    float* rpre  = qpre  + (size_t)N_QUERY * ATTN_DIM;        // 401*64
    float* htau  = rpre  + (size_t)N_REL_TOT * ATTN_DIM;      // 731*128
    float* htpre = htau  + (size_t)N_DT * IN_DIM;             // 731*64

    hipMemsetAsync(msg, 0, (size_t)N_NODE * IN_DIM * sizeof(float), stream);

    qpre_kernel <<<N_QUERY,   ATTN_DIM, 0, stream>>>(q_rel, rela, WqrW, Wqrb, qpre);
    rpre_kernel <<<N_REL_TOT, ATTN_DIM, 0, stream>>>(rela, Wr, rpre);
    htau_kernel <<<N_DT,      IN_DIM,   0, stream>>>(wt1, bt1, wt2, bt2, htau);
    htpre_kernel<<<N_DT,      ATTN_DIM, 0, stream>>>(htau, Wtau, htpre);

    int edge_blocks = (N_EDGE / 16 + 7) / 8;   // 31250 tiles, 8 waves/block
    edge_kernel<<<edge_blocks, 256, 0, stream>>>(hidden, edges, q_tau, rela, Ws,
                                                 waW, wab, qpre, rpre, htau, htpre, msg);

    int gemm_blocks = (N_NODE / 16 + 7) / 8;   // 3125 strips
    out_gemm<<<gemm_blocks, 256, 0, stream>>>(msg, Wh, (float*)d_out);
}